// E71DeltaCell_54906861912664
// MI455X (gfx1250) — compile-verified
//
#include <hip/hip_runtime.h>

typedef __attribute__((ext_vector_type(2))) float v2f;
typedef __attribute__((ext_vector_type(8))) float v8f;

#define TT 2048
#define BB 16
#define DD 1024
#define NN 128
#define MM (TT * BB)
#define EPSV 1e-6f

// ---------------------------------------------------------------------------
// Kernel 1: fused input projections  O[m,n] = sum_d X[m,d] * W[n,d]
// f32 WMMA (V_WMMA_F32_16X16X4_F32). grid (MM/128, 4), block 256 (8 waves).
// blockIdx.y selects which of the 4 weight matrices / output buffers.
// ---------------------------------------------------------------------------
__global__ __launch_bounds__(256)
void proj_gemm_kernel(const float* __restrict__ X,
                      const float* __restrict__ Wk, const float* __restrict__ Wv,
                      const float* __restrict__ Wq, const float* __restrict__ Wb,
                      float* __restrict__ Ok, float* __restrict__ Ov,
                      float* __restrict__ Oq, float* __restrict__ Ob)
{
    const int which = blockIdx.y;
    const float* __restrict__ W = (which == 0) ? Wk : (which == 1) ? Wv : (which == 2) ? Wq : Wb;
    float* __restrict__ O       = (which == 0) ? Ok : (which == 1) ? Ov : (which == 2) ? Oq : Ob;

    const int tid  = threadIdx.x;
    const int lane = tid & 31;          // wave32
    const int wave = tid >> 5;          // 0..7, each owns a 16-row M strip
    const int m0   = blockIdx.x * 128;

    __shared__ float Xs[128][33];       // pitch 33 -> no systematic bank conflicts
    __shared__ float Ws[128][33];

    v8f acc[8];                          // 8 N-tiles of 16 -> full N=128
    #pragma unroll
    for (int i = 0; i < 8; i++) { v8f z = {}; acc[i] = z; }

    const int lrow = lane & 15;          // row (A) / col (B) within tile
    const int klo  = (lane >> 4) * 2;    // K pair selector per ISA f32 layout

    // cooperative LDS fill: each thread moves 16 contiguous floats per array
    const int ldrow = tid >> 1;          // 0..127
    const int ldcol = (tid & 1) * 16;    // 0 or 16

    for (int kc = 0; kc < DD; kc += 32) {
        __syncthreads();
        {
            const float* xg = X + (size_t)(m0 + ldrow) * DD + (kc + ldcol);
            const float* wg = W + (size_t)ldrow * DD + (kc + ldcol);
            float4 xv[4], wv[4];
            #pragma unroll
            for (int u = 0; u < 4; u++) xv[u] = ((const float4*)xg)[u];
            #pragma unroll
            for (int u = 0; u < 4; u++) wv[u] = ((const float4*)wg)[u];
            #pragma unroll
            for (int u = 0; u < 4; u++) {
                Xs[ldrow][ldcol + 4*u + 0] = xv[u].x;
                Xs[ldrow][ldcol + 4*u + 1] = xv[u].y;
                Xs[ldrow][ldcol + 4*u + 2] = xv[u].z;
                Xs[ldrow][ldcol + 4*u + 3] = xv[u].w;
                Ws[ldrow][ldcol + 4*u + 0] = wv[u].x;
                Ws[ldrow][ldcol + 4*u + 1] = wv[u].y;
                Ws[ldrow][ldcol + 4*u + 2] = wv[u].z;
                Ws[ldrow][ldcol + 4*u + 3] = wv[u].w;
            }
        }
        __syncthreads();

        #pragma unroll
        for (int kk = 0; kk < 32; kk += 4) {
            v2f a;
            a[0] = Xs[wave * 16 + lrow][kk + klo];
            a[1] = Xs[wave * 16 + lrow][kk + klo + 1];
            #pragma unroll
            for (int nt = 0; nt < 8; nt++) {
                v2f bf;
                bf[0] = Ws[nt * 16 + lrow][kk + klo];
                bf[1] = Ws[nt * 16 + lrow][kk + klo + 1];
                acc[nt] = __builtin_amdgcn_wmma_f32_16x16x4_f32(
                    /*neg_a=*/false, a, /*neg_b=*/false, bf,
                    /*c_mod=*/(short)0, acc[nt],
                    /*reuse_a=*/false, /*reuse_b=*/false);
            }
        }
    }

    // Epilogue: C/D layout -> lane L holds col (L&15); VGPR r holds row r (+8 for L>=16)
    const int rbase = (lane >> 4) * 8;
    #pragma unroll
    for (int nt = 0; nt < 8; nt++) {
        #pragma unroll
        for (int r = 0; r < 8; r++) {
            const int m = m0 + wave * 16 + rbase + r;
            const int n = nt * 16 + lrow;
            O[(size_t)m * NN + n] = acc[nt][r];
        }
    }
}

// ---------------------------------------------------------------------------
// Kernel 2: sequential delta-rule scan. grid = B (16), block = 128.
// Thread i keeps S-row i (128 f32) in VGPRs; k/q broadcast through LDS.
// ---------------------------------------------------------------------------
__global__ __launch_bounds__(128)
void delta_scan_kernel(const float* __restrict__ kb, const float* __restrict__ vb,
                       const float* __restrict__ qb, const float* __restrict__ bxb,
                       const float* __restrict__ S0,
                       const float* __restrict__ d_beta, const float* __restrict__ b_beta,
                       float* __restrict__ out)
{
    const int i = threadIdx.x;   // state row 0..127
    const int b = blockIdx.x;    // batch

    __shared__ __align__(16) float ks[NN];
    __shared__ __align__(16) float qs[NN];

    float Srow[NN];
    #pragma unroll
    for (int j = 0; j < NN; j += 4) {
        float4 s = *(const float4*)&S0[((size_t)b * NN + i) * NN + j];
        Srow[j] = s.x; Srow[j + 1] = s.y; Srow[j + 2] = s.z; Srow[j + 3] = s.w;
    }

    const float db  = d_beta[i];
    const float bb0 = b_beta[i];

    for (int t = 0; t < TT; ++t) {
        const size_t base = ((size_t)t * BB + b) * NN;
        const float ki  = kb[base + i];
        const float qi  = qb[base + i];
        const float vi  = vb[base + i];
        const float bxi = bxb[base + i];

        __syncthreads();              // previous step's readers are done
        ks[i] = ki; qs[i] = qi;
        __syncthreads();

        // pass 1: ||k||^2 and raw dot S_row . k in one sweep
        float ss = 0.f, rd = 0.f;
        #pragma unroll
        for (int j = 0; j < NN; j += 4) {
            float4 kv = *(const float4*)&ks[j];
            ss = fmaf(kv.x, kv.x, fmaf(kv.y, kv.y, fmaf(kv.z, kv.z, fmaf(kv.w, kv.w, ss))));
            rd = fmaf(Srow[j], kv.x, fmaf(Srow[j+1], kv.y,
                 fmaf(Srow[j+2], kv.z, fmaf(Srow[j+3], kv.w, rd))));
        }
        const float inv  = 1.0f / (sqrtf(ss) + EPSV);
        const float retr = rd * inv;                       // S @ k_norm
        const float beta = 1.0f / (1.0f + __expf(-(fmaf(db, retr, bxi) + bb0)));
        const float c    = beta * (vi - retr) * inv;       // scale for rank-1 update

        // pass 2: fused rank-1 update + (updated S) @ q
        float o = 0.f;
        #pragma unroll
        for (int j = 0; j < NN; j += 4) {
            float4 kv = *(const float4*)&ks[j];
            float4 qv = *(const float4*)&qs[j];
            Srow[j]   = fmaf(c, kv.x, Srow[j]);
            Srow[j+1] = fmaf(c, kv.y, Srow[j+1]);
            Srow[j+2] = fmaf(c, kv.z, Srow[j+2]);
            Srow[j+3] = fmaf(c, kv.w, Srow[j+3]);
            o = fmaf(Srow[j], qv.x, fmaf(Srow[j+1], qv.y,
                fmaf(Srow[j+2], qv.z, fmaf(Srow[j+3], qv.w, o))));
        }
        const float sig = 1.0f / (1.0f + __expf(-o));
        out[base + i] = o * o * sig;                       // o * silu(o)
    }

    // final state, appended after outputs (tuple return order)
    float* Sf = out + (size_t)TT * BB * NN;
    #pragma unroll
    for (int j = 0; j < NN; j += 4) {
        *(float4*)&Sf[((size_t)b * NN + i) * NN + j] =
            make_float4(Srow[j], Srow[j+1], Srow[j+2], Srow[j+3]);
    }
}

// ---------------------------------------------------------------------------
extern "C" void kernel_launch(void* const* d_in, const int* in_sizes, int n_in,
                              void* d_out, int out_size, void* d_ws, size_t ws_size,
                              hipStream_t stream) {
    (void)in_sizes; (void)n_in; (void)out_size; (void)ws_size;
    const float* x     = (const float*)d_in[0];
    const float* S0    = (const float*)d_in[1];
    const float* Wk    = (const float*)d_in[2];
    const float* Wv    = (const float*)d_in[3];
    const float* Wq    = (const float*)d_in[4];
    const float* Wb    = (const float*)d_in[5];
    const float* dbeta = (const float*)d_in[6];
    const float* bbeta = (const float*)d_in[7];
    float* out = (float*)d_out;
    float* ws  = (float*)d_ws;

    const size_t per = (size_t)MM * NN;          // 4 x 16.8 MB activation buffers
    float* kb  = ws;
    float* vb  = ws + per;
    float* qb  = ws + 2 * per;
    float* bxb = ws + 3 * per;

    dim3 g1(MM / 128, 4);
    proj_gemm_kernel<<<g1, 256, 0, stream>>>(x, Wk, Wv, Wq, Wb, kb, vb, qb, bxb);
    delta_scan_kernel<<<BB, 128, 0, stream>>>(kb, vb, qb, bxb, S0, dbeta, bbeta, out);
}